// GraphConvolution_24550033064495
// MI455X (gfx1250) — compile-verified
//
#include <hip/hip_runtime.h>

typedef __attribute__((ext_vector_type(2))) float v2f;
typedef __attribute__((ext_vector_type(8))) float v8f;

#define B_   4
#define NV_  2048
#define NE_  4096
#define FVI  64
#define FVO  64
#define FEI  32

#define NT   128      // n-rows per block
#define MT   64       // m-cols per tile
#define KC   32       // K chunk staged in LDS
#define MSPLIT 4      // m-range split across blocks
#define MRANGE (NV_ / MSPLIT)

#define ASTR 34       // padded LDS strides (8B-aligned, bank-conflict friendly)
#define BSTR 34
#define PSTR 66

// ---------------------------------------------------------------- prep: d = H_e @ p, copy H_e to output tail
__global__ void k_prep(const float* __restrict__ He, const float* __restrict__ p,
                       float* __restrict__ dbuf, float* __restrict__ He_out) {
    int idx = blockIdx.x * blockDim.x + threadIdx.x;       // one thread per (b,e)
    if (idx >= B_ * NE_) return;
    const float* row = He + (size_t)idx * FEI;
    float acc = 0.f;
#pragma unroll
    for (int f = 0; f < FEI; ++f) acc += row[f] * p[f];
    dbuf[idx] = acc;
    const float4* src = (const float4*)row;
    float4* dst = (float4*)(He_out + (size_t)idx * FEI);
#pragma unroll
    for (int q = 0; q < FEI / 4; ++q) dst[q] = src[q];
}

// ---------------------------------------------------------------- HW = H_v @ W
__global__ void k_hw(const float* __restrict__ Hv, const float* __restrict__ W,
                     float* __restrict__ HW) {
    int row = blockIdx.x;            // b*NV + n
    int o   = threadIdx.x;           // 0..63
    const float* hv = Hv + (size_t)row * FVI;
    float acc = 0.f;
#pragma unroll 8
    for (int f = 0; f < FVI; ++f) acc += hv[f] * W[f * FVO + o];
    HW[(size_t)row * FVO + o] = acc;
}

// ---------------------------------------------------------------- out pre-fill with bias (atomics accumulate on top)
__global__ void k_bias(const float* __restrict__ bias, float* __restrict__ out) {
    int idx = blockIdx.x * blockDim.x + threadIdx.x;
    if (idx >= B_ * NV_ * FVO) return;
    out[idx] = bias[idx & (FVO - 1)];
}

// ---------------------------------------------------------------- fused: mult tile -> modulate -> second GEMM
__global__ void __launch_bounds__(256)
k_main(const float* __restrict__ T, const float* __restrict__ adj_v,
       const float* __restrict__ dbuf, const float* __restrict__ HW,
       float* __restrict__ out) {
    __shared__ float As[NT * ASTR];   // Td rows (n-slice), K chunk
    __shared__ float Bs[MT * BSTR];   // T rows (m-slice),  K chunk
    __shared__ float Pb[NT * PSTR];   // modulated mult tile, wave-private row strips

    const int b  = blockIdx.z;
    const int mc = blockIdx.y;
    const int n0 = blockIdx.x * NT;
    const int tid  = threadIdx.x;
    const int w    = tid >> 5;        // wave 0..7
    const int lane = tid & 31;
    const int lr   = lane & 15;       // row/col within 16x16 tile
    const int lh   = lane >> 4;       // K-pair selector

    const float* Tb  = T     + (size_t)b * NV_ * NE_;
    const float* Ab  = adj_v + (size_t)b * NV_ * NV_;
    const float* db  = dbuf  + (size_t)b * NE_;
    const float* HWb = HW    + (size_t)b * NV_ * FVO;
    float*       ob  = out   + (size_t)b * NV_ * FVO;

    const v8f zero = {};
    v8f Oacc[4];
#pragma unroll
    for (int t = 0; t < 4; ++t) Oacc[t] = zero;

    for (int mi = 0; mi < MRANGE / MT; ++mi) {
        const int m0 = mc * MRANGE + mi * MT;

        v8f Pacc[4];
#pragma unroll
        for (int t = 0; t < 4; ++t) Pacc[t] = zero;

        // ---- phase 1: Pacc = (T*d)[n-slice] @ T[m-slice]^T over K = NE
        for (int kc = 0; kc < NE_; kc += KC) {
            __syncthreads();
            // stage As = T[n0+row][kc+*] * d[kc+*]   (128 x 32)
#pragma unroll
            for (int i = 0; i < (NT * KC / 4) / 256; ++i) {      // 4 float4 / thread
                int idx4 = tid + i * 256;
                int row  = idx4 >> 3;
                int cg   = (idx4 & 7) << 2;
                float4 tv = *(const float4*)&Tb[(size_t)(n0 + row) * NE_ + kc + cg];
                float4 dv = *(const float4*)&db[kc + cg];
                float* d4 = &As[row * ASTR + cg];
                d4[0] = tv.x * dv.x; d4[1] = tv.y * dv.y;
                d4[2] = tv.z * dv.z; d4[3] = tv.w * dv.w;
            }
            // stage Bs = T[m0+row][kc+*]              (64 x 32)
#pragma unroll
            for (int i = 0; i < (MT * KC / 4) / 256; ++i) {      // 2 float4 / thread
                int idx4 = tid + i * 256;
                int row  = idx4 >> 3;
                int cg   = (idx4 & 7) << 2;
                float4 tv = *(const float4*)&Tb[(size_t)(m0 + row) * NE_ + kc + cg];
                float* d4 = &Bs[row * BSTR + cg];
                d4[0] = tv.x; d4[1] = tv.y; d4[2] = tv.z; d4[3] = tv.w;
            }
            __syncthreads();

#pragma unroll
            for (int kk = 0; kk < KC; kk += 4) {
                const v2f a = *(const v2f*)&As[(16 * w + lr) * ASTR + kk + 2 * lh];
#pragma unroll
                for (int t = 0; t < 4; ++t) {
                    const v2f bb = *(const v2f*)&Bs[(16 * t + lr) * BSTR + kk + 2 * lh];
                    Pacc[t] = __builtin_amdgcn_wmma_f32_16x16x4_f32(
                        false, a, false, bb, (short)0, Pacc[t], false, false);
                }
            }
        }

        // ---- phase 2: modulate with eye/adj_v, park in LDS (wave-private rows)
#pragma unroll
        for (int t = 0; t < 4; ++t) {
#pragma unroll
            for (int r = 0; r < 8; ++r) {
                int rn = 16 * w + r + 8 * lh;               // local row 0..127
                int gn = n0 + rn;
                int gm = m0 + 16 * t + lr;
                float av  = Ab[(size_t)gn * NV_ + gm];
                float val = (gn == gm) ? av : Pacc[t][r] * av;   // M1*adj_v
                Pb[rn * PSTR + 16 * t + lr] = val;
            }
        }
        // DS ops are in-order per wave and rows are wave-private: no barrier needed.

        // ---- phase 3: Oacc += Padj(16x64) @ HW[m-slice](64x64), K = MT
#pragma unroll
        for (int kk = 0; kk < MT; kk += 4) {
            const v2f a = *(const v2f*)&Pb[(16 * w + lr) * PSTR + kk + 2 * lh];
            const float* hb = &HWb[(size_t)(m0 + kk + 2 * lh) * FVO];
#pragma unroll
            for (int t = 0; t < 4; ++t) {
                v2f bb;
                bb.x = hb[16 * t + lr];
                bb.y = hb[FVO + 16 * t + lr];
                Oacc[t] = __builtin_amdgcn_wmma_f32_16x16x4_f32(
                    false, a, false, bb, (short)0, Oacc[t], false, false);
            }
        }
    }

    // ---- epilogue: combine m-splits into bias-initialized out
#pragma unroll
    for (int t = 0; t < 4; ++t) {
#pragma unroll
        for (int r = 0; r < 8; ++r) {
            int gn  = n0 + 16 * w + r + 8 * lh;
            int col = 16 * t + lr;
            atomicAdd(&ob[(size_t)gn * FVO + col], Oacc[t][r]);
        }
    }
}

// ---------------------------------------------------------------- host
extern "C" void kernel_launch(void* const* d_in, const int* in_sizes, int n_in,
                              void* d_out, int out_size, void* d_ws, size_t ws_size,
                              hipStream_t stream) {
    const float* H_v    = (const float*)d_in[0];
    const float* H_e    = (const float*)d_in[1];
    // d_in[2] = adj_e (unused by node_layer path)
    const float* adj_v  = (const float*)d_in[3];
    const float* T      = (const float*)d_in[4];
    const float* weight = (const float*)d_in[5];
    const float* p      = (const float*)d_in[6];
    const float* bias   = (const float*)d_in[7];

    float* out    = (float*)d_out;                    // [B,NV,FVO]
    float* He_out = out + (size_t)B_ * NV_ * FVO;     // [B,NE,FEI]

    float* dbuf = (float*)d_ws;                       // [B,NE]
    float* HWb  = dbuf + (size_t)B_ * NE_;            // [B,NV,FVO]

    k_prep<<<(B_ * NE_ + 255) / 256, 256, 0, stream>>>(H_e, p, dbuf, He_out);
    k_hw<<<B_ * NV_, FVO, 0, stream>>>(H_v, weight, HWb);
    k_bias<<<(B_ * NV_ * FVO + 255) / 256, 256, 0, stream>>>(bias, out);

    dim3 grid(NV_ / NT, MSPLIT, B_);
    k_main<<<grid, 256, 0, stream>>>(T, adj_v, dbuf, HWb, out);
}